// AttnWrapper_36258113913468
// MI455X (gfx1250) — compile-verified
//
#include <hip/hip_runtime.h>

#define DIM      768
#define NLINES   8192
#define OVL      32           // WINDOW_HALF * GAP = 8*4
#define NHEADS   8
#define HD       96
#define NWIN     16

typedef __attribute__((ext_vector_type(2))) float v2f;
typedef __attribute__((ext_vector_type(8))) float v8f;

// Row gather over the virtual "combined = [begin; main; end]" tensor.
__device__ __forceinline__ const float* a_row(const float* __restrict__ beg,
                                              const float* __restrict__ mid,
                                              const float* __restrict__ end,
                                              int nbeg, int nmain, int r) {
  if (r < nbeg) return beg + (size_t)r * DIM;
  r -= nbeg;
  if (r < nmain) return mid + (size_t)r * DIM;
  return end + (size_t)(r - nmain) * DIM;
}

// Out[m, out_col0+n] = (sum_k A[m,k] * Wt[n,k] + bias[n]) * scale
// One 16x64 tile per wave (4 N-subtiles share one A fragment):
// per k-step = 1 A load + 4 B loads -> 4 WMMAs (fp32 16x16x4).
__global__ __launch_bounds__(256) void gemm_nt_wmma(
    const float* __restrict__ Abeg, const float* __restrict__ Amain,
    const float* __restrict__ Aend, int nbeg, int nmain,
    const float* __restrict__ Wt, const float* __restrict__ bias,
    float* __restrict__ Out, int out_stride, int out_col0,
    float scale, int mtiles, int ntiles64)
{
  const int lane = threadIdx.x & 31;
  const int wave = threadIdx.x >> 5;
  const int t    = blockIdx.x * 8 + wave;
  if (t >= mtiles * ntiles64) return;            // wave-uniform exit

  const int mt = t / ntiles64, nt = t % ntiles64;
  const int m0 = mt << 4,      n0 = nt << 6;     // 16 rows x 64 cols
  const int half = lane >> 4,  lr = lane & 15;

  // A fragment: lanes 0-15 hold K = k+0,k+1 of row m0+lr; lanes 16-31 hold K = k+2,k+3.
  const float* ap = a_row(Abeg, Amain, Aend, nbeg, nmain, m0 + lr);
  // B fragments (Wt row-major [outdim, K]; B[k][n] = Wt[n][k]); j-th subtile col = n0+16j+lr.
  const float* bp = Wt + (size_t)(n0 + lr) * DIM;

  v8f acc0 = {}, acc1 = {}, acc2 = {}, acc3 = {};
  #pragma unroll 4
  for (int k = 0; k < DIM; k += 4) {
    const int kk = k + (half << 1);
    v2f a, b0, b1, b2, b3;
    a.x  = ap[kk];                  a.y  = ap[kk + 1];
    b0.x = bp[kk];                  b0.y = bp[kk + 1];
    b1.x = bp[kk + 16 * DIM];       b1.y = bp[kk + 16 * DIM + 1];
    b2.x = bp[kk + 32 * DIM];       b2.y = bp[kk + 32 * DIM + 1];
    b3.x = bp[kk + 48 * DIM];       b3.y = bp[kk + 48 * DIM + 1];
    acc0 = __builtin_amdgcn_wmma_f32_16x16x4_f32(false, a, false, b0, (short)0, acc0, false, false);
    acc1 = __builtin_amdgcn_wmma_f32_16x16x4_f32(false, a, false, b1, (short)0, acc1, false, false);
    acc2 = __builtin_amdgcn_wmma_f32_16x16x4_f32(false, a, false, b2, (short)0, acc2, false, false);
    acc3 = __builtin_amdgcn_wmma_f32_16x16x4_f32(false, a, false, b3, (short)0, acc3, false, false);
  }

  // C/D layout: vgpr i, lane l -> row = i + 8*(l>=16), col = l%16
  const float bv0 = bias[n0 + lr];
  const float bv1 = bias[n0 + 16 + lr];
  const float bv2 = bias[n0 + 32 + lr];
  const float bv3 = bias[n0 + 48 + lr];
  #pragma unroll
  for (int i = 0; i < 8; i++) {
    const size_t r = (size_t)(m0 + i + (half << 3));
    float* o = Out + r * out_stride + out_col0 + n0 + lr;
    o[0]  = (acc0[i] + bv0) * scale;
    o[16] = (acc1[i] + bv1) * scale;
    o[32] = (acc2[i] + bv2) * scale;
    o[48] = (acc3[i] + bv3) * scale;
  }
}

// One wave per (line, head). Lanes split hd=96 into 3 elems each.
__global__ __launch_bounds__(256) void attn_win(
    const float* __restrict__ Q,   // [NLINES, DIM], already scaled by hd^-0.5
    const float* __restrict__ K,   // [NLINES+2*OVL, DIM]  (combined rows)
    const float* __restrict__ V,
    float* __restrict__ CTX)       // [NLINES, DIM]
{
  const int n    = blockIdx.x;
  const int h    = threadIdx.x >> 5;
  const int lane = threadIdx.x & 31;
  const int d0   = lane * 3;

  const float* q = Q + (size_t)n * DIM + h * HD;
  const float q0 = q[d0], q1 = q[d0 + 1], q2 = q[d0 + 2];

  float sc[NWIN];
  #pragma unroll
  for (int w = 0; w < NWIN; w++) {
    const int off  = (w < 8) ? (-OVL + w * 4) : (4 + (w - 8) * 4);
    const float* kp = K + (size_t)(n + OVL + off) * DIM + h * HD;
    float p = q0 * kp[d0] + q1 * kp[d0 + 1] + q2 * kp[d0 + 2];
    p += __shfl_xor(p, 16, 32);
    p += __shfl_xor(p,  8, 32);
    p += __shfl_xor(p,  4, 32);
    p += __shfl_xor(p,  2, 32);
    p += __shfl_xor(p,  1, 32);
    sc[w] = p;                      // all lanes hold full dot product
  }

  float mx = sc[0];
  #pragma unroll
  for (int w = 1; w < NWIN; w++) mx = fmaxf(mx, sc[w]);
  float sum = 0.f;
  #pragma unroll
  for (int w = 0; w < NWIN; w++) { sc[w] = __expf(sc[w] - mx); sum += sc[w]; }
  const float inv = 1.f / sum;

  float c0 = 0.f, c1 = 0.f, c2 = 0.f;
  #pragma unroll
  for (int w = 0; w < NWIN; w++) {
    const int off  = (w < 8) ? (-OVL + w * 4) : (4 + (w - 8) * 4);
    const float* vp = V + (size_t)(n + OVL + off) * DIM + h * HD;
    const float a = sc[w] * inv;
    c0 += a * vp[d0]; c1 += a * vp[d0 + 1]; c2 += a * vp[d0 + 2];
  }
  float* o = CTX + (size_t)n * DIM + h * HD;
  o[d0] = c0; o[d0 + 1] = c1; o[d0 + 2] = c2;
}

// out[:, 0:768] = main  (out row stride = 1536)
__global__ __launch_bounds__(256) void copy_main_half(
    const float4* __restrict__ src, float* __restrict__ out)
{
  const size_t i = (size_t)blockIdx.x * blockDim.x + threadIdx.x;
  const size_t r = i / (DIM / 4);
  const size_t c = i % (DIM / 4);
  if (r < NLINES)
    *reinterpret_cast<float4*>(out + r * (2 * DIM) + c * 4) = src[i];
}

extern "C" void kernel_launch(void* const* d_in, const int* in_sizes, int n_in,
                              void* d_out, int out_size, void* d_ws, size_t ws_size,
                              hipStream_t stream) {
  const float* mainp  = (const float*)d_in[0];
  const float* beginp = (const float*)d_in[1];
  const float* endp   = (const float*)d_in[2];
  const float* in_w   = (const float*)d_in[3];   // [3*768, 768]
  const float* in_b   = (const float*)d_in[4];   // [3*768]
  const float* out_w  = (const float*)d_in[5];   // [768, 768]
  const float* out_b  = (const float*)d_in[6];   // [768]
  float* out = (float*)d_out;                    // [NLINES, 1536]

  float* ws  = (float*)d_ws;
  float* Q   = ws;                                         // NLINES*DIM
  float* Kb  = Q  + (size_t)NLINES * DIM;                  // (NLINES+64)*DIM
  float* Vb  = Kb + (size_t)(NLINES + 2 * OVL) * DIM;      // (NLINES+64)*DIM
  float* CTX = Vb + (size_t)(NLINES + 2 * OVL) * DIM;      // NLINES*DIM

  const float* wq = in_w;
  const float* wk = in_w + (size_t)DIM * DIM;
  const float* wv = in_w + (size_t)2 * DIM * DIM;
  const float* bq = in_b;
  const float* bk = in_b + DIM;
  const float* bv = in_b + 2 * DIM;

  const float qscale = 0.10206207261596577f;   // 96^-0.5
  const int nt64 = DIM / 64;                   // 12
  const int mtQ  = NLINES / 16;                // 512
  const int mtKV = (NLINES + 2 * OVL) / 16;    // 516

  // Q = (main @ wq.T + bq) * scale
  gemm_nt_wmma<<<(mtQ * nt64 + 7) / 8, 256, 0, stream>>>(
      mainp, mainp, mainp, 0, NLINES, wq, bq, Q, DIM, 0, qscale, mtQ, nt64);
  // K = combined @ wk.T + bk   (gather begin/main/end inline)
  gemm_nt_wmma<<<(mtKV * nt64 + 7) / 8, 256, 0, stream>>>(
      beginp, mainp, endp, OVL, NLINES, wk, bk, Kb, DIM, 0, 1.0f, mtKV, nt64);
  // V = combined @ wv.T + bv
  gemm_nt_wmma<<<(mtKV * nt64 + 7) / 8, 256, 0, stream>>>(
      beginp, mainp, endp, OVL, NLINES, wv, bv, Vb, DIM, 0, 1.0f, mtKV, nt64);

  // windowed attention -> CTX
  attn_win<<<NLINES, 256, 0, stream>>>(Q, Kb, Vb, CTX);

  // out[:, :768] = main
  copy_main_half<<<(NLINES * (DIM / 4) + 255) / 256, 256, 0, stream>>>(
      (const float4*)mainp, out);

  // out[:, 768:] = CTX @ out_w.T + out_b
  gemm_nt_wmma<<<(mtQ * nt64 + 7) / 8, 256, 0, stream>>>(
      CTX, CTX, CTX, 0, NLINES, out_w, out_b, out, 2 * DIM, DIM, 1.0f, mtQ, nt64);
}